// GraphSAGE_20950850470131
// MI455X (gfx1250) — compile-verified
//
#include <hip/hip_runtime.h>

// GraphSAGE (2-layer, mean agg) + top-1 prompt routing + per-cluster heads.
// Dense GEMMs use CDNA5 V_WMMA_F32_16X16X4_F32 (exact fp32 matrix pipe).

typedef __attribute__((ext_vector_type(2))) float v2f;
typedef __attribute__((ext_vector_type(8))) float v8f;

#define N_FEAT    128   // IN_FEATS == HIDDEN == 128
#define N_CLASSES 40
#define CENTERS   16

// ---------------------------------------------------------------- zero fill
__global__ void zero_f4(float4* __restrict__ p, int n4) {
    int i = blockIdx.x * blockDim.x + threadIdx.x;
    if (i < n4) p[i] = make_float4(0.f, 0.f, 0.f, 0.f);
}

// ------------------------------------------------- edge scatter (segment sum)
// One wave32 per edge; each lane moves 4 contiguous features (32*4 = 128).
__global__ void __launch_bounds__(256)
scatter_edges(const float* __restrict__ feat,
              const int*   __restrict__ src,
              const int*   __restrict__ dst,
              float* __restrict__ agg,
              float* __restrict__ deg,
              int nEdges, int addDeg) {
    int t    = blockIdx.x * blockDim.x + threadIdx.x;
    int e    = t >> 5;
    int lane = t & 31;
    if (e >= nEdges) return;
    int s = src[e];
    int d = dst[e];
    float4 v = ((const float4*)(feat + (size_t)s * N_FEAT))[lane];
    float* base = agg + (size_t)d * N_FEAT + lane * 4;
    atomicAdd(base + 0, v.x);
    atomicAdd(base + 1, v.y);
    atomicAdd(base + 2, v.z);
    atomicAdd(base + 3, v.w);
    if (addDeg && lane == 0) atomicAdd(deg + d, 1.0f);
}

// --------------------------------------------------------- fused SAGE layer
// out = relu( (agg/max(deg,1)) @ wl^T + bl + xin @ wr^T )
// Block = 256 threads = 8 waves; wave w owns output columns [16w,16w+16),
// block b owns node rows [16b, 16b+16). K=128 consumed as 32 x (K=4) WMMAs
// per GEMM, both GEMMs chained into one f32 accumulator.
__global__ void __launch_bounds__(256)
sage_gemm(const float* __restrict__ agg, const float* __restrict__ deg,
          const float* __restrict__ xin, const float* __restrict__ wl,
          const float* __restrict__ bl,  const float* __restrict__ wr,
          float* __restrict__ out) {
    const int lane    = threadIdx.x & 31;
    const int wave    = threadIdx.x >> 5;
    const int l16     = lane & 15;
    const int half    = lane >> 4;
    const int rowBase = blockIdx.x * 16;
    const int colBase = wave * 16;
    const int r = rowBase + l16;            // A-matrix row held by this lane
    const int n = colBase + l16;            // B-matrix column held by this lane

    const float invd = 1.0f / fmaxf(deg[r], 1.0f);
    const float* aggRow = agg + (size_t)r * N_FEAT;
    const float* xRow   = xin + (size_t)r * N_FEAT;
    const float* wlRow  = wl  + (size_t)n * N_FEAT;   // w[n][k] contiguous in k
    const float* wrRow  = wr  + (size_t)n * N_FEAT;

    v8f c = {};
#pragma unroll 4
    for (int k0 = 0; k0 < N_FEAT; k0 += 4) {
        const int ko = k0 + 2 * half;       // lane's K pair within the K=4 step
        v2f am = *(const v2f*)(aggRow + ko);
        am *= invd;                          // fold mean normalization into A
        v2f ax = *(const v2f*)(xRow + ko);
        v2f bL = *(const v2f*)(wlRow + ko);
        v2f bR = *(const v2f*)(wrRow + ko);
        c = __builtin_amdgcn_wmma_f32_16x16x4_f32(false, am, false, bL, (short)0, c, false, false);
        c = __builtin_amdgcn_wmma_f32_16x16x4_f32(false, ax, false, bR, (short)0, c, false, false);
    }

    const float bias = bl[n];
#pragma unroll
    for (int m = 0; m < 8; ++m) {
        const int row = rowBase + half * 8 + m;   // C/D layout: VGPR m, lane-half
        out[(size_t)row * N_FEAT + n] = fmaxf(c[m] + bias, 0.0f);
    }
}

// --------------------------------- routing: logits (WMMA) + per-row argmax
// One wave per 16-node tile: logits tile = h[16x128] @ prompt_w^T[128x16].
__global__ void __launch_bounds__(256)
route_argmax(const float* __restrict__ h, const float* __restrict__ pw,
             int* __restrict__ idxOut, int nTiles) {
    const int lane = threadIdx.x & 31;
    const int wave = threadIdx.x >> 5;
    const int tile = blockIdx.x * (blockDim.x >> 5) + wave;
    if (tile >= nTiles) return;               // wave-uniform guard (EXEC all-1)
    const int l16     = lane & 15;
    const int half    = lane >> 4;
    const int rowBase = tile * 16;

    const float* hRow = h  + (size_t)(rowBase + l16) * N_FEAT;
    const float* pRow = pw + (size_t)l16 * N_FEAT;   // center l16, contiguous in k

    v8f c = {};
#pragma unroll 4
    for (int k0 = 0; k0 < N_FEAT; k0 += 4) {
        const int ko = k0 + 2 * half;
        v2f a = *(const v2f*)(hRow + ko);
        v2f b = *(const v2f*)(pRow + ko);
        c = __builtin_amdgcn_wmma_f32_16x16x4_f32(false, a, false, b, (short)0, c, false, false);
    }

    // Row m of each lane-half lives in c[m] across 16 lanes (lane = center id).
    // Reduce (max, first-index) within each 16-lane half.
#pragma unroll
    for (int m = 0; m < 8; ++m) {
        float val = c[m];
        int   best = l16;
#pragma unroll
        for (int s = 8; s >= 1; s >>= 1) {
            float ov = __shfl_xor(val, s, 32);
            int   oi = __shfl_xor(best, s, 32);
            if (ov > val || (ov == val && oi < best)) { val = ov; best = oi; }
        }
        if (l16 == 0) idxOut[rowBase + half * 8 + m] = best;
    }
}

// ------------------------------------- selected-expert class head (40 dots)
__global__ void __launch_bounds__(256)
expert_head(const float* __restrict__ h, const int* __restrict__ idx,
            const float* __restrict__ ppw, float* __restrict__ out, int nNodes) {
    const int lane = threadIdx.x & 31;
    const int wave = threadIdx.x >> 5;
    const int node = blockIdx.x * (blockDim.x >> 5) + wave;
    if (node >= nNodes) return;
    const float4 hv = ((const float4*)(h + (size_t)node * N_FEAT))[lane];
    const int k = idx[node];
    const float* W = ppw + (size_t)k * N_CLASSES * N_FEAT;
    for (int cc = 0; cc < N_CLASSES; ++cc) {
        const float4 wv = ((const float4*)(W + (size_t)cc * N_FEAT))[lane];
        float p = hv.x * wv.x + hv.y * wv.y + hv.z * wv.z + hv.w * wv.w;
#pragma unroll
        for (int s = 16; s >= 1; s >>= 1) p += __shfl_xor(p, s, 32);
        if (lane == 0) out[(size_t)node * N_CLASSES + cc] = p;
    }
}

// --------------------------------------------------------------------------
extern "C" void kernel_launch(void* const* d_in, const int* in_sizes, int n_in,
                              void* d_out, int out_size, void* d_ws, size_t ws_size,
                              hipStream_t stream) {
    const float* x        = (const float*)d_in[0];
    const int*   edges    = (const int*)d_in[1];   // [2, E] int32
    const float* w_l0     = (const float*)d_in[2];
    const float* b_l0     = (const float*)d_in[3];
    const float* w_r0     = (const float*)d_in[4];
    const float* w_l1     = (const float*)d_in[5];
    const float* b_l1     = (const float*)d_in[6];
    const float* w_r1     = (const float*)d_in[7];
    const float* prompt_w = (const float*)d_in[8];
    const float* pp_w     = (const float*)d_in[9];

    const int N = in_sizes[0] / N_FEAT;   // 40000
    const int E = in_sizes[1] / 2;        // 640000
    const int* src = edges;
    const int* dst = edges + E;

    // workspace layout (agg immediately followed by deg so one zero pass covers both)
    float* agg = (float*)d_ws;                       // N*128 f32
    float* deg = agg + (size_t)N * N_FEAT;           // N    f32
    float* h0  = deg + N;                            // N*128 f32
    float* h1  = h0 + (size_t)N * N_FEAT;            // N*128 f32
    int*   idx = (int*)(h1 + (size_t)N * N_FEAT);    // N    i32
    float* outp = (float*)d_out;

    const int scatterBlocks = (int)(((long long)E * 32 + 255) / 256);

    // ---- layer 0
    {
        int n4 = (N * N_FEAT + N) / 4;               // agg + deg together
        zero_f4<<<(n4 + 255) / 256, 256, 0, stream>>>((float4*)agg, n4);
        scatter_edges<<<scatterBlocks, 256, 0, stream>>>(x, src, dst, agg, deg, E, 1);
        sage_gemm<<<N / 16, 256, 0, stream>>>(agg, deg, x, w_l0, b_l0, w_r0, h0);
    }
    // ---- layer 1 (deg is reused; only agg re-zeroed)
    {
        int n4 = (N * N_FEAT) / 4;
        zero_f4<<<(n4 + 255) / 256, 256, 0, stream>>>((float4*)agg, n4);
        scatter_edges<<<scatterBlocks, 256, 0, stream>>>(h0, src, dst, agg, deg, E, 0);
        sage_gemm<<<N / 16, 256, 0, stream>>>(agg, deg, h0, w_l1, b_l1, w_r1, h1);
    }
    // ---- routing + expert heads
    {
        int nTiles = N / 16;
        route_argmax<<<(nTiles + 7) / 8, 256, 0, stream>>>(h1, prompt_w, idx, nTiles);
        expert_head<<<(N + 7) / 8, 256, 0, stream>>>(h1, idx, pp_w, outp, N);
    }
}